// FNOnd_35218731827947
// MI455X (gfx1250) — compile-verified
//
#include <hip/hip_runtime.h>
#include <hip/hip_bf16.h>
#include <math.h>

// ---------------------------------------------------------------------------
// FNO on MI455X (gfx1250): all FFTs truncated to 16x16 modes -> dense GEMMs
// executed with V_WMMA_F32_16X16X4_F32. One wave32 per (16 x multi-16) output
// tile; A fragments reused across N-tiles for higher WMMA/byte.
// ---------------------------------------------------------------------------

typedef float v2f __attribute__((ext_vector_type(2)));
typedef float v8f __attribute__((ext_vector_type(8)));

#define HH 256
#define WW 256
#define HW 65536
#define NIMG 128            // B * CH
#define IMG_STRIDE 65536
#define CHAN_BLOCK 2097152  // 32*65536, one batch's channel block

static __device__ __forceinline__ v8f wmma4(v2f a, v2f b, v8f c) {
  // D = A(16x4,f32) * B(4x16,f32) + C(16x16,f32)
  return __builtin_amdgcn_wmma_f32_16x16x4_f32(
      /*neg_a=*/false, a, /*neg_b=*/false, b,
      /*c_mod=*/(short)0, c, /*reuse_a=*/false, /*reuse_b=*/false);
}

static __device__ __forceinline__ float gelu_exact(float x) {
  return x * 0.5f * (1.0f + erff(x * 0.70710678118654752440f));
}

// ---------------------------------------------------------------------------
// Init: precompute DFT basis matrices (orthonormal split into 1/16 per axis)
//   w1p    : B-packed forward-W basis  [kp=128][n=32] float2
//   a2     : forward-H basis A operand [32][512] float (plain, contiguous K)
//   ebuf   : inverse-H basis A operand [256][32] float
//   binvp  : B-packed inverse-W basis  [kp=16][256] float2
//   bbwp   : B-packed bypass weights   [app=16][kp=16][n=32] float2
// ---------------------------------------------------------------------------
__global__ void k_init(float* __restrict__ w1p, float* __restrict__ a2,
                       float* __restrict__ ebuf, float* __restrict__ binvp,
                       float* __restrict__ bbwp, const float* __restrict__ byp_w) {
  const int tid = blockIdx.x * 256 + threadIdx.x; // [0,16384)
  const float c = 6.28318530717958647692f / 256.0f;
  const float inv16 = 1.0f / 16.0f;

  { // A2 [32][512]: rows 0..15 = Re(k), 16..31 = Im(k); cols 0..255 over u_re, 256..511 over u_im
    int m = tid >> 9, k = tid & 511;
    int kr = m & 15, imrow = m >> 4;
    int h = k & 255, compcol = k >> 8;
    float ang = c * (float)(kr * h);
    float Chv = cosf(ang) * inv16, Shv = sinf(ang) * inv16;
    float v = imrow ? (compcol ? Chv : -Shv) : (compcol ? Shv : Chv);
    a2[tid] = v;
  }
  if (tid < 8192) { // E [256][32]: cols 0..15 cos, 16..31 sin
    int h = tid >> 5, cc = tid & 31;
    int kk = cc & 15;
    float ang = c * (float)(kk * h);
    ebuf[tid] = (cc < 16 ? cosf(ang) : sinf(ang)) * inv16;
  }
  if (tid < 4096) { // W1pack: forward W basis, B role. cols 0..15: cos, 16..31: -sin
    int kp = tid >> 5, n = tid & 31;
    int w0 = 2 * kp, w1 = 2 * kp + 1;
    float v0, v1;
    if (n < 16) { v0 = cosf(c * (float)(n * w0)) * inv16;
                  v1 = cosf(c * (float)(n * w1)) * inv16; }
    else        { int l = n - 16;
                  v0 = -sinf(c * (float)(l * w0)) * inv16;
                  v1 = -sinf(c * (float)(l * w1)) * inv16; }
    v2f* P = (v2f*)w1p; v2f v; v.x = v0; v.y = v1; P[tid] = v;
  }
  if (tid < 4096) { // Binvpack: inverse W basis [l'=32][w=256], pair-packed over l'
    int kp = tid >> 8, w = tid & 255;
    float vv[2];
    for (int s = 0; s < 2; ++s) {
      int lp = 2 * kp + s;
      float v;
      if (lp < 16) { int l = lp; v = ((l == 0) ? 1.0f : 2.0f) * cosf(c * (float)(l * w)) * inv16; }
      else { int l = lp - 16; v = (l == 0) ? 0.0f : (-2.0f * inv16) * sinf(c * (float)(l * w)); }
      vv[s] = v;
    }
    v2f* P = (v2f*)binvp; v2f v; v.x = vv[0]; v.y = vv[1]; P[kp * 256 + w] = v;
  }
  if (tid < 8192) { // Bbwpack: bypass weights transposed (B[k=i][n=o]), pair-packed over i
    int app = tid >> 9, rest = tid & 511;
    int kp = rest >> 5, n = rest & 31;
    const float* bw = byp_w + app * 1024; // [o][i]
    v2f v; v.x = bw[n * 32 + 2 * kp]; v.y = bw[n * 32 + 2 * kp + 1];
    ((v2f*)bbwp)[tid] = v;
  }
}

// ---------------------------------------------------------------------------
// Timestep embedding + 2-layer MLP (tiny). temb[b*32+j]
// ---------------------------------------------------------------------------
__global__ __launch_bounds__(128) void k_temb(const int* __restrict__ t,
                                              const float* __restrict__ w1, const float* __restrict__ b1,
                                              const float* __restrict__ w2, const float* __restrict__ b2,
                                              float* __restrict__ temb) {
  __shared__ float emb[4][32];
  __shared__ float h1[4][32];
  const int tid = threadIdx.x;
  const int b = tid >> 5, j = tid & 31;
  float tv = (float)t[b];
  const float k = -9.21034037197618274f / 15.0f; // -ln(10000)/(half-1)
  if (j < 16) emb[b][j] = sinf(tv * expf(k * (float)j));
  else        emb[b][j] = cosf(tv * expf(k * (float)(j - 16)));
  __syncthreads();
  float s = b1[j];
  for (int i = 0; i < 32; ++i) s += emb[b][i] * w1[j * 32 + i];
  h1[b][j] = gelu_exact(s);
  __syncthreads();
  float s2 = b2[j];
  for (int i = 0; i < 32; ++i) s2 += h1[b][i] * w2[j * 32 + i];
  temb[b * 32 + j] = s2;
}

// ---------------------------------------------------------------------------
// Lift: x0[b,ch,h,w] = lw[ch,0]*x + lw[ch,1]*c + lb[ch] + temb[b,ch]
// ---------------------------------------------------------------------------
__global__ void k_lift(const float* __restrict__ x, const float* __restrict__ cc,
                       const float* __restrict__ lw, const float* __restrict__ lb,
                       const float* __restrict__ temb, float* __restrict__ x0) {
  const int tid = blockIdx.x * 256 + threadIdx.x; // 8388608 total
  const int b = tid >> 21;
  const int ch = (tid >> 16) & 31;
  const int pix = tid & 65535;
  const float xv = x[b * HW + pix];
  const float cv = cc[b * HW + pix];
  x0[tid] = lw[ch * 2 + 0] * xv + lw[ch * 2 + 1] * cv + lb[ch] + temb[b * 32 + ch];
}

// ---------------------------------------------------------------------------
// Forward DFT stage 1: per image U = X(256x256) @ W1(256x32).
// Each wave computes a 16x32 tile (both complex components), reusing A.
// Output pair-packed (K-pairs) for stage-2 B operand:
//   upack[img][comp][hp=128][l=16] float2
// Grid: (128 imgs, 16 Mtiles), 32 threads.
// ---------------------------------------------------------------------------
__global__ __launch_bounds__(32) void k_fwd1(const float* __restrict__ xb,
                                             const float* __restrict__ w1p,
                                             float* __restrict__ upack) {
  const int img = blockIdx.x;
  const int m0 = blockIdx.y << 4;
  const int lane = threadIdx.x & 31;
  const int l15 = lane & 15, hi = lane >> 4;
  const float* A = xb + img * IMG_STRIDE + (m0 + l15) * 256 + 2 * hi;
  const v2f* Bp = (const v2f*)w1p;
  v8f acc0 = {}, acc1 = {};
#pragma unroll 8
  for (int k0 = 0; k0 < 256; k0 += 4) {
    v2f a = *(const v2f*)(A + k0);
    const int kp = (k0 >> 1) + hi;
    v2f b0 = Bp[kp * 32 + l15];
    v2f b1 = Bp[kp * 32 + 16 + l15];
    acc0 = wmma4(a, b0, acc0);
    acc1 = wmma4(a, b1, acc1);
  }
  v2f* U = (v2f*)upack + img * 4096 + l15;
#pragma unroll
  for (int p = 0; p < 4; ++p) {
    int hp = (m0 >> 1) + p + 4 * hi;
    v2f v0; v0.x = acc0[2 * p]; v0.y = acc0[2 * p + 1];
    v2f v1; v1.x = acc1[2 * p]; v1.y = acc1[2 * p + 1];
    U[hp * 16] = v0;            // comp 0 (u_re)
    U[2048 + hp * 16] = v1;     // comp 1 (u_im)
  }
}

// ---------------------------------------------------------------------------
// Forward DFT stage 2: XF(32x16) = A2(32x512) @ V(512x16), V = packed U.
// XF rows 0..15 = Re over k, 16..31 = Im over k; cols = l.
// Grid: (128 imgs, 2 Mtiles), 32 threads.
// ---------------------------------------------------------------------------
__global__ __launch_bounds__(32) void k_fwd2(const float* __restrict__ a2,
                                             const float* __restrict__ upack,
                                             float* __restrict__ xf) {
  const int img = blockIdx.x;
  const int m0 = blockIdx.y << 4;
  const int lane = threadIdx.x & 31;
  const int l15 = lane & 15, hi = lane >> 4;
  const float* A = a2 + (m0 + l15) * 512 + 2 * hi;
  const v2f* Bp = (const v2f*)upack + img * 4096 + l15;
  v8f acc = {};
#pragma unroll 8
  for (int k0 = 0; k0 < 512; k0 += 4) {
    v2f a = *(const v2f*)(A + k0);
    v2f b = Bp[((k0 >> 1) + hi) * 16];
    acc = wmma4(a, b, acc);
  }
  float* D = xf + img * 512;
#pragma unroll
  for (int r = 0; r < 8; ++r) D[(m0 + r + 8 * hi) * 16 + l15] = acc[r];
}

// ---------------------------------------------------------------------------
// Spectral channel mix: per mode (k,l): of[b,o] = sum_i xf[b,i]*(wr+i*wi)[i,o]
// One block per k covers ALL batches so each weight element is loaded once.
// Writes OF in the duplicated real layout for inverse stage 1:
//   OF[k][l]=re  OF[k][16+l]=im  OF[16+k][l]=-im  OF[16+k][16+l]=re
// Grid: (16 k), 512 threads (o=tid>>4, l=tid&15).
// ---------------------------------------------------------------------------
__global__ __launch_bounds__(512) void k_spec(const float* __restrict__ xf,
                                              const float* __restrict__ wr,
                                              const float* __restrict__ wi,
                                              float* __restrict__ of, int app) {
  const int k = blockIdx.x;
  const int tid = threadIdx.x;
  const int o = tid >> 4, l = tid & 15;
  __shared__ float xr[4][32][16];
  __shared__ float xi[4][32][16];
#pragma unroll
  for (int bb = 0; bb < 4; ++bb) {
    const float* X = xf + (bb * 32 + o) * 512; // o doubles as channel i for the load
    xr[bb][o][l] = X[k * 16 + l];
    xi[bb][o][l] = X[(16 + k) * 16 + l];
  }
  __syncthreads();
  const long base = (long)app * 262144 + (long)o * 256 + k * 16 + l;
  const float* WR = wr + base;
  const float* WI = wi + base;
  float ore[4] = {0.f, 0.f, 0.f, 0.f};
  float oim[4] = {0.f, 0.f, 0.f, 0.f};
#pragma unroll 4
  for (int i = 0; i < 32; ++i) {
    float wre = WR[i * 8192], wim = WI[i * 8192];
#pragma unroll
    for (int bb = 0; bb < 4; ++bb) {
      float xre = xr[bb][i][l], xim = xi[bb][i][l];
      ore[bb] = fmaf(xre, wre, ore[bb]); ore[bb] = fmaf(-xim, wim, ore[bb]);
      oim[bb] = fmaf(xre, wim, oim[bb]); oim[bb] = fmaf(xim, wre, oim[bb]);
    }
  }
#pragma unroll
  for (int bb = 0; bb < 4; ++bb) {
    float* OF = of + (bb * 32 + o) * 1024;
    OF[k * 32 + l] = ore[bb];
    OF[k * 32 + 16 + l] = oim[bb];
    OF[(16 + k) * 32 + l] = -oim[bb];
    OF[(16 + k) * 32 + 16 + l] = ore[bb];
  }
}

// ---------------------------------------------------------------------------
// Inverse stage 1: Z(256x32) = E(256x32) @ OF(32x32), per output image.
// Each wave computes 16x32 (both halves), reusing A. Z cols 0..15 Re, 16..31 Im.
// Grid: (128 imgs, 16 Mtiles), 32 threads.
// ---------------------------------------------------------------------------
__global__ __launch_bounds__(32) void k_inv1(const float* __restrict__ ebuf,
                                             const float* __restrict__ of,
                                             float* __restrict__ z) {
  const int img = blockIdx.x;
  const int m0 = blockIdx.y << 4;
  const int lane = threadIdx.x & 31;
  const int l15 = lane & 15, hi = lane >> 4;
  const float* A = ebuf + (m0 + l15) * 32 + 2 * hi;
  const float* B = of + img * 1024;
  v8f acc0 = {}, acc1 = {};
#pragma unroll
  for (int k0 = 0; k0 < 32; k0 += 4) {
    v2f a = *(const v2f*)(A + k0);
    const int r0 = (k0 + 2 * hi) * 32;
    v2f b0, b1;
    b0.x = B[r0 + l15];           b0.y = B[r0 + 32 + l15];
    b1.x = B[r0 + 16 + l15];      b1.y = B[r0 + 48 + l15];
    acc0 = wmma4(a, b0, acc0);
    acc1 = wmma4(a, b1, acc1);
  }
  float* Z = z + img * 8192;
#pragma unroll
  for (int r = 0; r < 8; ++r) {
    const int row = (m0 + r + 8 * hi) * 32;
    Z[row + l15] = acc0[r];
    Z[row + 16 + l15] = acc1[r];
  }
}

// ---------------------------------------------------------------------------
// Bypass conv1x1 as GEMM over pixels: tile M=16 pixels (along w), N=32 out
// channels (A reused), K=32 in channels. Writes (byp + bias) to xout.
// Grid: (1024 = b*256+h, 16 wtiles), 32 threads.
// ---------------------------------------------------------------------------
__global__ __launch_bounds__(32) void k_byp(const float* __restrict__ xin,
                                            const float* __restrict__ bbwp,
                                            const float* __restrict__ bb,
                                            float* __restrict__ xout, int app) {
  const int bh = blockIdx.x;
  const int b = bh >> 8, h = bh & 255;
  const int w0 = blockIdx.y << 4;
  const int lane = threadIdx.x & 31;
  const int l15 = lane & 15, hi = lane >> 4;
  const float* A = xin + b * CHAN_BLOCK + hi * (2 * IMG_STRIDE) + h * 256 + w0 + l15;
  const v2f* Bp = (const v2f*)bbwp + app * 512;
  v8f acc0 = {}, acc1 = {};
#pragma unroll
  for (int k0 = 0; k0 < 32; k0 += 4) {
    v2f a;
    a.x = A[k0 * IMG_STRIDE];
    a.y = A[k0 * IMG_STRIDE + IMG_STRIDE];
    const int kp = (k0 >> 1) + hi;
    v2f b0 = Bp[kp * 32 + l15];
    v2f b1 = Bp[kp * 32 + 16 + l15];
    acc0 = wmma4(a, b0, acc0);
    acc1 = wmma4(a, b1, acc1);
  }
  const float bias0 = bb[app * 32 + l15];
  const float bias1 = bb[app * 32 + 16 + l15];
  float* O0 = xout + b * CHAN_BLOCK + l15 * IMG_STRIDE + h * 256 + w0 + 8 * hi;
  float* O1 = O0 + 16 * IMG_STRIDE;
#pragma unroll
  for (int r = 0; r < 8; ++r) {
    O0[r] = acc0[r] + bias0;
    O1[r] = acc1[r] + bias1;
  }
}

// ---------------------------------------------------------------------------
// Inverse stage 2 + residual-add + GELU: spec(256x256) = Z(256x32) @ Binv(32x256)
// then xout = gelu(spec + xout) in place. Each wave computes a 16h x 64w tile
// (4 N-tiles, A reused 4x). Grid: (128 imgs, 16 htiles, 4 wquads), 32 threads.
// ---------------------------------------------------------------------------
__global__ __launch_bounds__(32) void k_inv2(const float* __restrict__ z,
                                             const float* __restrict__ binvp,
                                             float* __restrict__ xout) {
  const int img = blockIdx.x;
  const int m0 = blockIdx.y << 4;
  const int n0 = blockIdx.z << 6;
  const int lane = threadIdx.x & 31;
  const int l15 = lane & 15, hi = lane >> 4;
  const float* A = z + img * 8192 + (m0 + l15) * 32 + 2 * hi;
  const v2f* Bp = (const v2f*)binvp + n0 + l15;
  v8f acc[4] = {{}, {}, {}, {}};
#pragma unroll
  for (int k0 = 0; k0 < 32; k0 += 4) {
    v2f a = *(const v2f*)(A + k0);
    const int kp = (k0 >> 1) + hi;
#pragma unroll
    for (int j = 0; j < 4; ++j) {
      v2f b = Bp[kp * 256 + j * 16];
      acc[j] = wmma4(a, b, acc[j]);
    }
  }
  float* O = xout + img * IMG_STRIDE + n0 + l15;
#pragma unroll
  for (int j = 0; j < 4; ++j) {
#pragma unroll
    for (int r = 0; r < 8; ++r) {
      int idx = (m0 + r + 8 * hi) * 256 + j * 16;
      O[idx] = gelu_exact(O[idx] + acc[j][r]);
    }
  }
}

// ---------------------------------------------------------------------------
// Projection: out[b,a,h,w] = sum_i pw[i]*xb[b,i,h,w] + pb
// ---------------------------------------------------------------------------
__global__ void k_proj(const float* __restrict__ xb, const float* __restrict__ pw,
                       const float* __restrict__ pb, float* __restrict__ out, int a) {
  const int tid = blockIdx.x * 256 + threadIdx.x; // 262144 total
  const int b = tid >> 16, pix = tid & 65535;
  const float* X = xb + b * CHAN_BLOCK + pix;
  float s = pb[0];
#pragma unroll 8
  for (int i = 0; i < 32; ++i) s = fmaf(pw[i], X[i * IMG_STRIDE], s);
  out[(b * 4 + a) * HW + pix] = s;
}

// ---------------------------------------------------------------------------
extern "C" void kernel_launch(void* const* d_in, const int* in_sizes, int n_in,
                              void* d_out, int out_size, void* d_ws, size_t ws_size,
                              hipStream_t stream) {
  (void)in_sizes; (void)n_in; (void)out_size; (void)ws_size;
  const float* x       = (const float*)d_in[0];
  const int*   t       = (const int*)  d_in[1];
  const float* c       = (const float*)d_in[2];
  const float* lift_w  = (const float*)d_in[3];
  const float* lift_b  = (const float*)d_in[4];
  const float* tm1_w   = (const float*)d_in[5];
  const float* tm1_b   = (const float*)d_in[6];
  const float* tm2_w   = (const float*)d_in[7];
  const float* tm2_b   = (const float*)d_in[8];
  const float* spec_wr = (const float*)d_in[9];
  const float* spec_wi = (const float*)d_in[10];
  const float* byp_w   = (const float*)d_in[11];
  const float* byp_b   = (const float*)d_in[12];
  const float* proj_w  = (const float*)d_in[13];
  const float* proj_b  = (const float*)d_in[14];
  float* out = (float*)d_out;

  float* ws = (float*)d_ws;
  size_t off = 0;
  float* x0    = ws + off; off += 8388608;  // (4,32,256,256)
  float* bufA  = ws + off; off += 8388608;
  float* bufB  = ws + off; off += 8388608;
  float* upack = ws + off; off += 1048576;  // 128 imgs * 4096 float2
  float* xf    = ws + off; off += 65536;    // 128 * 32*16
  float* of    = ws + off; off += 131072;   // 128 * 32*32
  float* zbuf  = ws + off; off += 1048576;  // 128 * 256*32
  float* temb  = ws + off; off += 128;
  float* w1p   = ws + off; off += 8192;     // 128*32 float2
  float* a2    = ws + off; off += 16384;    // 32*512
  float* ebuf  = ws + off; off += 8192;     // 256*32
  float* binvp = ws + off; off += 8192;     // 16*256 float2
  float* bbwp  = ws + off; off += 16384;    // 16*16*32 float2

  k_init<<<64, 256, 0, stream>>>(w1p, a2, ebuf, binvp, bbwp, byp_w);
  k_temb<<<1, 128, 0, stream>>>(t, tm1_w, tm1_b, tm2_w, tm2_b, temb);
  k_lift<<<32768, 256, 0, stream>>>(x, c, lift_w, lift_b, temb, x0);

  float* bufs[2] = {bufA, bufB};
  for (int a = 0; a < 4; ++a) {
    const float* cur = x0;
    for (int blk = 0; blk < 4; ++blk) {
      const int app = a * 4 + blk;
      float* nxt = bufs[blk & 1];
      k_fwd1<<<dim3(NIMG, 16),     32, 0, stream>>>(cur, w1p, upack);
      k_fwd2<<<dim3(NIMG, 2),      32, 0, stream>>>(a2, upack, xf);
      k_spec<<<dim3(16),          512, 0, stream>>>(xf, spec_wr, spec_wi, of, app);
      k_inv1<<<dim3(NIMG, 16),     32, 0, stream>>>(ebuf, of, zbuf);
      k_byp <<<dim3(1024, 16),     32, 0, stream>>>(cur, bbwp, byp_b, nxt, app);
      k_inv2<<<dim3(NIMG, 16, 4),  32, 0, stream>>>(zbuf, binvp, nxt);
      cur = nxt;
    }
    k_proj<<<1024, 256, 0, stream>>>(cur, proj_w, proj_b, out, a);
  }
}